// Word2DM_57088705298719
// MI455X (gfx1250) — compile-verified
//
#include <hip/hip_runtime.h>
#include <math.h>

typedef __attribute__((ext_vector_type(2))) float v2f;
typedef __attribute__((ext_vector_type(8))) float v8f;

#define NDIM 20
#define EMB  (NDIM * NDIM)   // 400 floats per embedding
#define LDST 32              // padded LDS row stride (floats)
#define KNEG 10
#define WPB  8               // waves per block

// Exact parameter types for __builtin_amdgcn_global_load_async_to_lds_b128:
// (v4i addrspace(1)*, v4i addrspace(3)*, imm offset, imm cpol)
typedef int v4i_ __attribute__((vector_size(16)));
typedef __attribute__((address_space(1))) v4i_  as1_v4i;
typedef __attribute__((address_space(3))) v4i_  as3_v4i;
typedef __attribute__((address_space(3))) float lfloat;

// D = A(16x4 f32) x B(4x16 f32) + C(16x16 f32)
__device__ __forceinline__ v8f wmma_f32_16x16x4(v2f a, v2f b, v8f c) {
  return __builtin_amdgcn_wmma_f32_16x16x4_f32(
      /*neg_a=*/false, a, /*neg_b=*/false, b,
      /*c_mod=*/(short)0, c, /*reuse_a=*/false, /*reuse_b=*/false);
}

// One 16B chunk: global -> LDS via async DMA (ASYNCcnt-tracked, no VGPR trip).
__device__ __forceinline__ void async_b128(const float* g, float* l) {
#if __has_builtin(__builtin_amdgcn_global_load_async_to_lds_b128)
  __builtin_amdgcn_global_load_async_to_lds_b128(
      (as1_v4i*)g, (as3_v4i*)l, 0, 0);
#else
  unsigned lofs = (unsigned)(size_t)(lfloat*)l;
  unsigned long long ga = (unsigned long long)g;
  asm volatile("global_load_async_to_lds_b128 %0, %1, off"
               :: "v"(lofs), "v"(ga) : "memory");
#endif
}

// Stage a 400-float embedding (row-major [n=20][m=20]) into padded LDS
// [n*LDST + m]. Every 4-float chunk lies inside one row (20 = 5*4), so the
// async b128 copies map 1:1 onto the padded layout. Issues exactly 4 async ops.
__device__ __forceinline__ void stage_async(const float* __restrict__ src,
                                            float* lbuf, int lane) {
#pragma unroll
  for (int it = 0; it < 4; ++it) {
    int i4 = it * 32 + lane;                 // 16B chunk index
    if (it < 3 || lane < 4) {                // 100 chunks total
      int i = i4 * 4;
      int n = i / NDIM;
      int m = i - n * NDIM;                  // m in {0,4,8,12,16}
      async_b128(src + i, lbuf + n * LDST + m);
    }
  }
}

__device__ __forceinline__ void wait_async_le4() {
  asm volatile("s_wait_asynccnt 0x4" ::: "memory");
}
__device__ __forceinline__ void wait_async_0() {
  asm volatile("s_wait_asynccnt 0x0" ::: "memory");
}
__device__ __forceinline__ void wait_ds_0() {
  asm volatile("s_wait_dscnt 0x0" ::: "memory");
}

__device__ __forceinline__ float softplus(float x) {
  return fmaxf(x, 0.0f) + log1pf(__expf(-fabsf(x)));
}

template <bool ATOMIC>
__global__ __launch_bounds__(32 * WPB)
void word2dm_loss_kernel(const float* __restrict__ Wt,
                         const float* __restrict__ Wc,
                         const int* __restrict__ tgt,
                         const int* __restrict__ ctx,
                         const int* __restrict__ neg,
                         float* __restrict__ out,
                         int batch) {
  __shared__ float lds_all[WPB][2][NDIM * LDST];
  const int wave = threadIdx.x >> 5;
  const int lane = threadIdx.x & 31;
  const int b = blockIdx.x * WPB + wave;
  if (b >= batch) return;                    // waves fully independent: no barriers
  float* buf0 = lds_all[wave][0];
  float* buf1 = lds_all[wave][1];

  // Zero pad columns [20,32) of both buffers once; WMMA fragments falling
  // outside the real 20x20 matrix read these zeros -> free M/N padding.
  for (int i = lane; i < NDIM * (LDST - NDIM); i += 32) {
    int n = i / (LDST - NDIM);
    int c = NDIM + (i - n * (LDST - NDIM));
    buf0[n * LDST + c] = 0.0f;
    buf1[n * LDST + c] = 0.0f;
  }

  const int khalf  = (lane >> 4) << 1;       // 0 or 2 (K sub-pair per half-wave)
  const int lane15 = lane & 15;

  // ---- B_t -> buf0 (async), then preload A = B_t^T fragments to registers.
  stage_async(Wt + (size_t)tgt[b] * EMB, buf0, lane);
  wait_async_0();

  v2f afrag[2][5];
#pragma unroll
  for (int ti = 0; ti < 2; ++ti) {
    int mcol = ti * 16 + lane15;             // m index; >=20 hits zero padding
#pragma unroll
    for (int ks = 0; ks < 5; ++ks) {
      int kk = ks * 4 + khalf;               // n (contraction) index, <= 19
      afrag[ti][ks].x = buf0[kk * LDST + mcol];
      afrag[ti][ks].y = buf0[(kk + 1) * LDST + mcol];
    }
  }
  wait_ds_0();                               // afrag reads done: buf0 reusable

  // ---- software pipeline: B_0 (context) in flight, then ping-pong prefetch.
  stage_async(Wc + (size_t)ctx[b] * EMB, buf1, lane);

  float loss = 0.0f;
#pragma unroll
  for (int j = 0; j <= KNEG; ++j) {
    float* cur = (j & 1) ? buf0 : buf1;
    float* nxt = (j & 1) ? buf1 : buf0;

    if (j < KNEG) {                          // prefetch B_{j+1} into nxt
      int nidx = neg[(size_t)b * KNEG + j];
      wait_ds_0();                           // WAR: prior reads of nxt complete
      stage_async(Wc + (size_t)nidx * EMB, nxt, lane);
      wait_async_le4();                      // current matrix landed; prefetch flies
    } else {
      wait_async_0();                        // last matrix landed
    }

    v8f acc[2][2] = {};
#pragma unroll
    for (int ks = 0; ks < 5; ++ks) {
      int kk = ks * 4 + khalf;
      v2f bfrag[2];
#pragma unroll
      for (int tj = 0; tj < 2; ++tj) {       // B[k][ncol]; ncol>=20 -> zeros
        bfrag[tj].x = cur[kk * LDST + tj * 16 + lane15];
        bfrag[tj].y = cur[(kk + 1) * LDST + tj * 16 + lane15];
      }
#pragma unroll
      for (int ti = 0; ti < 2; ++ti)
#pragma unroll
        for (int tj = 0; tj < 2; ++tj)
          acc[ti][tj] = wmma_f32_16x16x4(afrag[ti][ks], bfrag[tj], acc[ti][tj]);
    }

    // squared Frobenius norm: per-lane partial, then wave32 xor-reduction
    float s = 0.0f;
#pragma unroll
    for (int ti = 0; ti < 2; ++ti)
#pragma unroll
      for (int tj = 0; tj < 2; ++tj)
#pragma unroll
        for (int e = 0; e < 8; ++e) {
          float c = acc[ti][tj][e];
          s += c * c;
        }
#pragma unroll
    for (int mask = 16; mask >= 1; mask >>= 1)
      s += __shfl_xor(s, mask, 32);

    // j==0: -log_sigmoid(s) = softplus(-s); negatives: softplus(s)
    loss += softplus(j == 0 ? -s : s);
  }

  if (lane == 0) {
    if (ATOMIC) atomicAdd(out, loss / (float)batch);
    else        out[b] = loss;
  }
}

__global__ __launch_bounds__(256)
void reduce_mean_kernel(const float* __restrict__ in, float* __restrict__ out,
                        int n) {
  __shared__ float sm[256];
  float s = 0.0f;
  for (int i = threadIdx.x; i < n; i += 256) s += in[i];
  sm[threadIdx.x] = s;
  __syncthreads();
  for (int w = 128; w > 0; w >>= 1) {
    if ((int)threadIdx.x < w) sm[threadIdx.x] += sm[threadIdx.x + w];
    __syncthreads();
  }
  if (threadIdx.x == 0) out[0] = sm[0] / (float)n;
}

__global__ void zero_kernel(float* p) { p[0] = 0.0f; }

extern "C" void kernel_launch(void* const* d_in, const int* in_sizes, int n_in,
                              void* d_out, int out_size, void* d_ws,
                              size_t ws_size, hipStream_t stream) {
  const float* Wt  = (const float*)d_in[0];
  const float* Wc  = (const float*)d_in[1];
  const int*   tix = (const int*)d_in[2];
  const int*   cix = (const int*)d_in[3];
  const int*   nix = (const int*)d_in[4];
  const int batch  = in_sizes[2];            // target_indices length
  const int blocks = (batch + WPB - 1) / WPB;

  if (ws_size >= (size_t)batch * sizeof(float)) {
    float* per_sample = (float*)d_ws;
    word2dm_loss_kernel<false><<<blocks, 32 * WPB, 0, stream>>>(
        Wt, Wc, tix, cix, nix, per_sample, batch);
    reduce_mean_kernel<<<1, 256, 0, stream>>>(per_sample, (float*)d_out, batch);
  } else {
    zero_kernel<<<1, 1, 0, stream>>>((float*)d_out);
    word2dm_loss_kernel<true><<<blocks, 32 * WPB, 0, stream>>>(
        Wt, Wc, tix, cix, nix, (float*)d_out, batch);
  }
}